// GCAETrainer_17935783428771
// MI455X (gfx1250) — compile-verified
//
#include <hip/hip_runtime.h>
#include <math.h>

#define Nn 3072
#define Dd 64
#define Bb 512
#define Ee 16
#define Hh 256
#define NBLK 48          // 3072 / 64
#define TB 64
#define TP (TB + 4)      // LDS row pitch: 68 floats = 272B (16B aligned rows, 4-bank skew)
#define NE 9             // K-1 edges per node
#define UNREACH_F 1.0e6f

typedef __attribute__((ext_vector_type(2))) float v2f;
typedef __attribute__((ext_vector_type(8))) float v8f;

// ---------------------------------------------------------------- async copy helpers (CDNA5)
// Per-lane async DMA: global -> LDS, 16B per lane per instruction, tracked by ASYNCcnt.
// LDS address = low 32 bits of the generic pointer to __shared__ (aperture layout).
__device__ __forceinline__ void async_tile_load64(const float* __restrict__ gbase,
                                                  int gstride, float* ldsbase, int tid) {
#pragma unroll
    for (int it = 0; it < 4; ++it) {
        int chunk = tid + (it << 8);          // 1024 chunks of 4 floats = 64x64 tile
        int rr = chunk >> 4;
        int cc = (chunk & 15) << 2;
        const float* g = gbase + (size_t)rr * gstride + cc;
        unsigned l = (unsigned)(uintptr_t)ldsbase + (unsigned)(((rr * TP) + cc) << 2);
        asm volatile("global_load_async_to_lds_b128 %0, %1, off"
                     :: "v"(l), "v"(g) : "memory");
    }
}
__device__ __forceinline__ void async_wait_all() {
    asm volatile("s_wait_asynccnt 0x0" ::: "memory");
}

// ---------------------------------------------------------------- scalars
__global__ void zero_scalars(float* s) {
    if (threadIdx.x == 0) { s[0] = 0.f; s[1] = 0.f; ((int*)s)[2] = 0; }
}

// ---------------------------------------------------------------- sq norms
__global__ void sqnorm_kernel(const float* __restrict__ X, float* __restrict__ sq) {
    int i = blockIdx.x * blockDim.x + threadIdx.x;
    if (i < Nn) {
        float s = 0.f;
        const float* r = X + (size_t)i * Dd;
        for (int d = 0; d < Dd; ++d) { float v = r[d]; s += v * v; }
        sq[i] = s;
    }
}

// ---------------------------------------------------------------- Gram (WMMA f32 16x16x4)
// d2[i][j] = max(0, sq[i] + sq[j] - 2 * dot(x_i, x_j))
__global__ void gram_wmma(const float* __restrict__ X, const float* __restrict__ sq,
                          float* __restrict__ d2) {
    int jt = blockIdx.x * 16;
    int it = blockIdx.y * 16;
    int lane = threadIdx.x & 31;
    int t = lane & 15, g = lane >> 4;
    const float* arow = X + (size_t)(it + t) * Dd;
    const float* brow = X + (size_t)(jt + t) * Dd;   // B[k][n] = X[jt+n][k]
    v8f acc = {};
    for (int k0 = 0; k0 < Dd; k0 += 4) {
        v2f a, b;
        a.x = arow[k0 + 2 * g];
        a.y = arow[k0 + 2 * g + 1];
        b.x = brow[k0 + 2 * g];
        b.y = brow[k0 + 2 * g + 1];
        acc = __builtin_amdgcn_wmma_f32_16x16x4_f32(false, a, false, b,
                                                    (short)0, acc, false, false);
    }
    for (int r = 0; r < 8; ++r) {
        int m = r + 8 * g, n = t;
        float v = sq[it + m] + sq[jt + n] - 2.0f * acc[r];
        d2[(size_t)(it + m) * Nn + (jt + n)] = fmaxf(v, 0.0f);
    }
}

// ---------------------------------------------------------------- KNN select (10 smallest, drop self)
__global__ void knn_select(float* __restrict__ d2, int* __restrict__ nidx,
                           float* __restrict__ nval) {
    int row = blockIdx.x;
    int tid = threadIdx.x;
    float* rw = d2 + (size_t)row * Nn;
    __shared__ float sval[256];
    __shared__ int   sidx[256];
    __shared__ float chosenV[10];
    __shared__ int   chosenI[10];
    for (int s = 0; s < 10; ++s) {
        float bv = 3.0e38f; int bidx = Nn;
        for (int c = tid; c < Nn; c += 256) {
            float v = rw[c];
            if (v < bv || (v == bv && c < bidx)) { bv = v; bidx = c; }
        }
        sval[tid] = bv; sidx[tid] = bidx;
        __syncthreads();
        for (int off = 128; off > 0; off >>= 1) {
            if (tid < off) {
                float ov = sval[tid + off]; int oi = sidx[tid + off];
                if (ov < sval[tid] || (ov == sval[tid] && oi < sidx[tid])) {
                    sval[tid] = ov; sidx[tid] = oi;
                }
            }
            __syncthreads();
        }
        if (tid == 0) {
            chosenV[s] = sval[0];
            chosenI[s] = sidx[0];
            rw[sidx[0]] = 3.0e38f;     // mask out for next pass
        }
        __syncthreads();
    }
    if (tid < NE) {
        int s = tid + 1;               // skip self (smallest)
        nidx[row * NE + tid] = chosenI[s];
        nval[row * NE + tid] = sqrtf(fmaxf(chosenV[s], 0.0f));
    }
}

// ---------------------------------------------------------------- graph init + edge scatter
__global__ void graph_init(float* __restrict__ Dm) {
    size_t idx = (size_t)blockIdx.x * blockDim.x + threadIdx.x;
    if (idx < (size_t)Nn * Nn) {
        int i = (int)(idx / Nn), j = (int)(idx % Nn);
        Dm[idx] = (i == j) ? 0.0f : UNREACH_F;
    }
}

__global__ void edge_scatter(const int* __restrict__ nidx, const float* __restrict__ nval,
                             float* __restrict__ Dm) {
    int e = blockIdx.x * blockDim.x + threadIdx.x;
    if (e < Nn * NE) {
        int i = e / NE;
        int c = nidx[e];
        float v = nval[e];
        if (v > 0.0f) {                 // mimics where(G > 0)
            Dm[(size_t)i * Nn + c] = v;
            Dm[(size_t)c * Nn + i] = v;
        }
    }
}

// ---------------------------------------------------------------- blocked Floyd-Warshall
// block = 16x16 threads, each owns a 4x4 sub-tile of a 64x64 tile
__global__ __launch_bounds__(256) void fw_phase1(float* __restrict__ Dm, int r) {
    __shared__ float T[TB][TP];
    int tx = threadIdx.x, ty = threadIdx.y;
    int tid = ty * 16 + tx;
    int base = r * TB;
    async_tile_load64(Dm + (size_t)base * Nn + base, Nn, &T[0][0], tid);
    async_wait_all();
    __syncthreads();
    for (int k = 0; k < TB; ++k) {
        for (int a = 0; a < 4; ++a)
            for (int b = 0; b < 4; ++b) {
                int i = ty + 16 * a, j = tx + 16 * b;
                T[i][j] = fminf(T[i][j], T[i][k] + T[k][j]);
            }
        __syncthreads();
    }
    for (int idx = tid; idx < TB * TB; idx += 256) {
        int rr = idx >> 6, cc = idx & 63;
        Dm[(size_t)(base + rr) * Nn + (base + cc)] = T[rr][cc];
    }
}

__global__ __launch_bounds__(256) void fw_phase2(float* __restrict__ Dm, int r) {
    int o = blockIdx.x;
    if (o == r) return;
    int isRow = (blockIdx.y == 0);
    __shared__ float P[TB][TP];
    __shared__ float C[TB][TP];
    int tx = threadIdx.x, ty = threadIdx.y;
    int tid = ty * 16 + tx;
    int pb = r * TB;
    int crow = isRow ? r * TB : o * TB;
    int ccol = isRow ? o * TB : r * TB;
    async_tile_load64(Dm + (size_t)pb * Nn + pb, Nn, &P[0][0], tid);
    async_tile_load64(Dm + (size_t)crow * Nn + ccol, Nn, &C[0][0], tid);
    async_wait_all();
    __syncthreads();
    for (int k = 0; k < TB; ++k) {
        if (isRow) {
            for (int a = 0; a < 4; ++a)
                for (int b = 0; b < 4; ++b) {
                    int i = ty + 16 * a, j = tx + 16 * b;
                    C[i][j] = fminf(C[i][j], P[i][k] + C[k][j]);
                }
        } else {
            for (int a = 0; a < 4; ++a)
                for (int b = 0; b < 4; ++b) {
                    int i = ty + 16 * a, j = tx + 16 * b;
                    C[i][j] = fminf(C[i][j], C[i][k] + P[k][j]);
                }
        }
        __syncthreads();
    }
    for (int idx = tid; idx < TB * TB; idx += 256) {
        int rr = idx >> 6, cc = idx & 63;
        Dm[(size_t)(crow + rr) * Nn + (ccol + cc)] = C[rr][cc];
    }
}

__global__ __launch_bounds__(256) void fw_phase3(float* __restrict__ Dm, int r) {
    int bx = blockIdx.x, by = blockIdx.y;
    if (bx == r || by == r) return;
    __shared__ float A[TB][TP];   // (by, r)
    __shared__ float B[TB][TP];   // (r, bx)
    int tx = threadIdx.x, ty = threadIdx.y;
    int tid = ty * 16 + tx;
    async_tile_load64(Dm + (size_t)(by * TB) * Nn + r * TB, Nn, &A[0][0], tid);
    async_tile_load64(Dm + (size_t)(r * TB) * Nn + bx * TB, Nn, &B[0][0], tid);
    float c[4][4];
    for (int a = 0; a < 4; ++a)
        for (int b = 0; b < 4; ++b)
            c[a][b] = Dm[(size_t)(by * TB + ty + 16 * a) * Nn + (bx * TB + tx + 16 * b)];
    async_wait_all();
    __syncthreads();
    for (int k = 0; k < TB; ++k) {
        float av[4], bv[4];
        for (int a = 0; a < 4; ++a) av[a] = A[ty + 16 * a][k];
        for (int b = 0; b < 4; ++b) bv[b] = B[k][tx + 16 * b];
        for (int a = 0; a < 4; ++a)
            for (int b = 0; b < 4; ++b)
                c[a][b] = fminf(c[a][b], av[a] + bv[b]);
    }
    for (int a = 0; a < 4; ++a)
        for (int b = 0; b < 4; ++b)
            Dm[(size_t)(by * TB + ty + 16 * a) * Nn + (bx * TB + tx + 16 * b)] = c[a][b];
}

// ---------------------------------------------------------------- finalize geo: threshold, write out, max
__global__ void geo_finalize(const float* __restrict__ Dm, float* __restrict__ out_geo,
                             float* __restrict__ scal) {
    __shared__ float smax[256];
    size_t idx = (size_t)blockIdx.x * blockDim.x + threadIdx.x;
    float v = 0.0f;
    if (idx < (size_t)Nn * Nn) {
        v = Dm[idx];
        v = (v >= UNREACH_F * 0.5f) ? 0.0f : v;
        out_geo[idx] = v;
    }
    smax[threadIdx.x] = v;
    __syncthreads();
    for (int off = 128; off > 0; off >>= 1) {
        if (threadIdx.x < off) smax[threadIdx.x] = fmaxf(smax[threadIdx.x], smax[threadIdx.x + off]);
        __syncthreads();
    }
    if (threadIdx.x == 0) atomicMax((int*)&scal[2], __float_as_int(smax[0]));
}

// ---------------------------------------------------------------- gather batch
__global__ void gather_batch(const float* __restrict__ X, const int* __restrict__ bi,
                             float* __restrict__ xb) {
    int idx = blockIdx.x * blockDim.x + threadIdx.x;
    if (idx < Bb * Dd) {
        int i = idx / Dd, d = idx % Dd;
        xb[idx] = X[(size_t)bi[i] * Dd + d];
    }
}

// ---------------------------------------------------------------- generic WMMA f32 GEMM, one wave / 16x16 tile
// C[m][n] = act(sum_k A[m][k]*B[k][n] + bias[n])
__global__ void gemm_wmma(const float* __restrict__ A, int lda,
                          const float* __restrict__ Bm, int ldb,
                          const float* __restrict__ bias,
                          float* __restrict__ C, int ldc,
                          int K, int do_relu) {
    int col = blockIdx.x * 16;
    int row = blockIdx.y * 16;
    int lane = threadIdx.x & 31;
    int t = lane & 15, g = lane >> 4;
    v8f acc = {};
    for (int k0 = 0; k0 < K; k0 += 4) {
        v2f a, b;
        a.x = A[(size_t)(row + t) * lda + k0 + 2 * g];
        a.y = A[(size_t)(row + t) * lda + k0 + 2 * g + 1];
        b.x = Bm[(size_t)(k0 + 2 * g) * ldb + col + t];
        b.y = Bm[(size_t)(k0 + 2 * g + 1) * ldb + col + t];
        acc = __builtin_amdgcn_wmma_f32_16x16x4_f32(false, a, false, b,
                                                    (short)0, acc, false, false);
    }
    for (int r = 0; r < 8; ++r) {
        int m = r + 8 * g, n = t;
        float v = acc[r] + bias[col + n];
        if (do_relu) v = fmaxf(v, 0.0f);
        C[(size_t)(row + m) * ldc + (col + n)] = v;
    }
}

// ---------------------------------------------------------------- reconstruction loss sum
__global__ void recon_loss(const float* __restrict__ xb, const float* __restrict__ xr,
                           float* __restrict__ scal) {
    __shared__ float ss[256];
    int idx = blockIdx.x * blockDim.x + threadIdx.x;
    float v = 0.0f;
    if (idx < Bb * Dd) { float d = xb[idx] - xr[idx]; v = d * d; }
    ss[threadIdx.x] = v;
    __syncthreads();
    for (int off = 128; off > 0; off >>= 1) {
        if (threadIdx.x < off) ss[threadIdx.x] += ss[threadIdx.x + off];
        __syncthreads();
    }
    if (threadIdx.x == 0) atomicAdd(&scal[0], ss[0]);
}

// ---------------------------------------------------------------- pairwise latent / geodesic loss sum
__global__ void pair_loss(const float* __restrict__ z, const int* __restrict__ bi,
                          const float* __restrict__ geo, float* __restrict__ scal) {
    __shared__ float ss[256];
    int idx = blockIdx.x * blockDim.x + threadIdx.x;
    float v = 0.0f;
    if (idx < Bb * Bb) {
        int i = idx >> 9, j = idx & (Bb - 1);
        float s = 0.0f;
        for (int e = 0; e < Ee; ++e) {
            float d = z[i * Ee + e] - z[j * Ee + e];
            s += d * d;
        }
        float zd = sqrtf(s + 1e-12f);
        float gb = geo[(size_t)bi[i] * Nn + bi[j]];
        float df = zd - gb;
        v = df * df;
    }
    ss[threadIdx.x] = v;
    __syncthreads();
    for (int off = 128; off > 0; off >>= 1) {
        if (threadIdx.x < off) ss[threadIdx.x] += ss[threadIdx.x + off];
        __syncthreads();
    }
    if (threadIdx.x == 0) atomicAdd(&scal[1], ss[0]);
}

// ---------------------------------------------------------------- final scalars
__global__ void final_losses(const float* __restrict__ scal, float* __restrict__ out) {
    if (threadIdx.x == 0) {
        float recon = scal[0] / (float)(Bb * Dd);
        float gmax = __int_as_float(((const int*)scal)[2]);
        float gl = (scal[1] / (float)(Bb * Bb)) / gmax;
        out[0] = recon + 1.0f * gl;   // ALPHA = 1.0
        out[1] = recon;
        out[2] = gl;
    }
}

// ================================================================ launch
extern "C" void kernel_launch(void* const* d_in, const int* in_sizes, int n_in,
                              void* d_out, int out_size, void* d_ws, size_t ws_size,
                              hipStream_t stream) {
    const float* x_train = (const float*)d_in[0];
    const int*   b_idx   = (const int*)d_in[1];
    const float* W_e1    = (const float*)d_in[2];
    const float* b_e1    = (const float*)d_in[3];
    const float* W_e2    = (const float*)d_in[4];
    const float* b_e2    = (const float*)d_in[5];
    const float* W_d1    = (const float*)d_in[6];
    const float* b_d1    = (const float*)d_in[7];
    const float* W_d2    = (const float*)d_in[8];
    const float* b_d2    = (const float*)d_in[9];
    float* out = (float*)d_out;

    // workspace layout (floats)
    float* ws   = (float*)d_ws;
    float* sq   = ws;                               // 3072 (+pad to 4096)
    float* Dm   = ws + 4096;                        // N*N
    float* nval = Dm + (size_t)Nn * Nn;             // N*NE
    int*   nidx = (int*)(nval + Nn * NE);           // N*NE ints
    float* xb   = (float*)(nidx + Nn * NE);         // B*D
    float* h    = xb + Bb * Dd;                     // B*H
    float* z    = h + Bb * Hh;                      // B*E
    float* hd   = z + Bb * Ee;                      // B*H
    float* xr   = hd + Bb * Hh;                     // B*D
    float* scal = xr + Bb * Dd;                     // 3: recon_sum, geo_sum, geo_max(bits)

    zero_scalars<<<1, 32, 0, stream>>>(scal);

    // ---- geodesic pipeline
    sqnorm_kernel<<<(Nn + 255) / 256, 256, 0, stream>>>(x_train, sq);
    gram_wmma<<<dim3(Nn / 16, Nn / 16), 32, 0, stream>>>(x_train, sq, Dm);
    knn_select<<<Nn, 256, 0, stream>>>(Dm, nidx, nval);
    {
        size_t total = (size_t)Nn * Nn;
        graph_init<<<(unsigned)((total + 255) / 256), 256, 0, stream>>>(Dm);
    }
    edge_scatter<<<(Nn * NE + 255) / 256, 256, 0, stream>>>(nidx, nval, Dm);

    for (int r = 0; r < NBLK; ++r) {
        fw_phase1<<<1, dim3(16, 16), 0, stream>>>(Dm, r);
        fw_phase2<<<dim3(NBLK, 2), dim3(16, 16), 0, stream>>>(Dm, r);
        fw_phase3<<<dim3(NBLK, NBLK), dim3(16, 16), 0, stream>>>(Dm, r);
    }

    {
        size_t total = (size_t)Nn * Nn;
        geo_finalize<<<(unsigned)((total + 255) / 256), 256, 0, stream>>>(Dm, out + 3, scal);
    }

    // ---- autoencoder forward (all WMMA)
    gather_batch<<<(Bb * Dd + 255) / 256, 256, 0, stream>>>(x_train, b_idx, xb);
    gemm_wmma<<<dim3(Hh / 16, Bb / 16), 32, 0, stream>>>(xb, Dd, W_e1, Hh, b_e1, h, Hh, Dd, 1);
    gemm_wmma<<<dim3(Ee / 16, Bb / 16), 32, 0, stream>>>(h, Hh, W_e2, Ee, b_e2, z, Ee, Hh, 0);
    gemm_wmma<<<dim3(Hh / 16, Bb / 16), 32, 0, stream>>>(z, Ee, W_d1, Hh, b_d1, hd, Hh, Ee, 1);
    gemm_wmma<<<dim3(Dd / 16, Bb / 16), 32, 0, stream>>>(hd, Hh, W_d2, Dd, b_d2, xr, Dd, Hh, 0);

    // ---- losses
    recon_loss<<<(Bb * Dd + 255) / 256, 256, 0, stream>>>(xb, xr, scal);
    pair_loss<<<(Bb * Bb + 255) / 256, 256, 0, stream>>>(z, b_idx, out + 3, scal);
    final_losses<<<1, 32, 0, stream>>>(scal, out);
}